// RejectionSamplerCompiled_22445499089621
// MI455X (gfx1250) — compile-verified
//
#include <hip/hip_runtime.h>
#include <stdint.h>

// CDNA5 / gfx1250 rejection sampler.
// One workgroup (512 threads = 16 wave32) per sequence.
// Pass A: stream target/draft prob rows through LDS via TDM (tensor_load_to_lds),
//         double-buffered, computing per-tile sums of adjusted = max(t-d,0) and t.
// CDF search: crossing tile found from <=13 tile sums; only that tile is re-read
//         (L2-hot) and block-scanned for the exact inverse-CDF index.

typedef __attribute__((ext_vector_type(4))) unsigned int v4u;
typedef __attribute__((ext_vector_type(8))) int          v8i;
typedef __attribute__((ext_vector_type(4))) int          v4i;

#define NTHREADS 512
#define NWAVES   (NTHREADS / 32)
#define TILE     4096
#define EPT      8      // TILE / NTHREADS
#define MAXTILES 32

// Issue one TDM descriptor: copy a contiguous run of `tileElems` f32 from
// global (zero-filled past `remaining` elements) into LDS at byte offset ldsOff.
// Descriptor packing per CDNA5 ISA (6-arg builtin).
__device__ __forceinline__ void tdm_load_tile(const float* gsrc, unsigned ldsOff,
                                              unsigned remaining, unsigned tileElems) {
  unsigned long long ga = (unsigned long long)(uintptr_t)gsrc;
  v4u g0;
  g0[0] = 1u;                                                 // count=1, user mode
  g0[1] = ldsOff;                                             // lds_addr (bytes)
  g0[2] = (unsigned)(ga & 0xffffffffu);                       // global_addr[31:0]
  g0[3] = (unsigned)((ga >> 32) & 0x01ffffffu) | (2u << 30);  // global_addr[56:32] | type=2
  v8i g1;
  g1[0] = (int)(2u << 16);                                    // data_size = 4 bytes
  g1[1] = (int)((remaining & 0xffffu) << 16);                 // tensor_dim0[15:0]
  g1[2] = (int)((remaining >> 16) | (1u << 16));              // tensor_dim0[31:16] | tensor_dim1=1
  g1[3] = (int)(tileElems << 16);                             // tile_dim0
  g1[4] = 1;                                                  // tile_dim1 = 1
  g1[5] = (int)remaining;                                     // tensor_dim0_stride
  g1[6] = 0;
  g1[7] = 0;
  v4i g2 = {0, 0, 0, 0};
  v4i g3 = {0, 0, 0, 0};
  v8i g4 = {0, 0, 0, 0, 0, 0, 0, 0};
  __builtin_amdgcn_tensor_load_to_lds(g0, g1, g2, g3, g4, 0);
}

__device__ __forceinline__ float wave_reduce_add(float x) {
  #pragma unroll
  for (int off = 16; off > 0; off >>= 1) x += __shfl_xor(x, off, 32);
  return x;
}

__device__ __forceinline__ int wave_reduce_add_i(int x) {
  #pragma unroll
  for (int off = 16; off > 0; off >>= 1) x += __shfl_xor(x, off, 32);
  return x;
}

__global__ __launch_bounds__(NTHREADS)
void rejection_sampler_kernel(const int*   __restrict__ draft_token_ids,   // [T]
                              const int*   __restrict__ num_draft_tokens,  // [B]
                              const float* __restrict__ draft_probs,       // [T,V]
                              const float* __restrict__ target_probs,      // [T,V]
                              const int*   __restrict__ bonus_token_ids,   // [B]
                              const float* __restrict__ uniform_samples,   // [T]
                              const float* __restrict__ resample_uniforms, // [B]
                              int* __restrict__ outbuf,                    // concatenated outputs
                              int B, int K, int V, int T) {
  __shared__ float bufT[2][TILE];
  __shared__ float bufD[2][TILE];
  __shared__ float tileA[MAXTILES];
  __shared__ float tileT[MAXTILES];
  __shared__ float sh_wsum[NWAVES];
  __shared__ float sh_woff[NWAVES];
  __shared__ int   shTok[32];
  __shared__ int   shAcc[32];
  __shared__ int   sh_meta[8];   // 0:start 1:num_acc 2:all_acc 3:rej_idx 4:cross 5:mode
  __shared__ float sh_fmeta[2];  // 0:thr 1:E(excl prefix at crossing tile)
  __shared__ int   sh_count;
  __shared__ int   sh_start;

  const int tid  = threadIdx.x;
  const int lane = tid & 31;
  const int wid  = tid >> 5;
  const int b    = blockIdx.x;

  // ---- start[b] = exclusive cumsum of num_draft_tokens ----
  if (tid == 0) { sh_start = 0; sh_count = 0; }
  __syncthreads();
  int contrib = (tid < b) ? num_draft_tokens[tid] : 0;   // b < B <= NTHREADS
  contrib = wave_reduce_add_i(contrib);
  if (lane == 0 && contrib != 0) atomicAdd(&sh_start, contrib);
  __syncthreads();
  const int start = sh_start;
  const int n_b   = num_draft_tokens[b];

  // ---- acceptance test for up to K draft tokens, in parallel ----
  if (tid < K) {
    int idx = start + tid;
    idx = idx < 0 ? 0 : (idx > T - 1 ? T - 1 : idx);
    const int dtok = draft_token_ids[idx];
    const size_t roff = (size_t)idx * (size_t)V + (size_t)dtok;
    float pd = fmaxf(draft_probs[roff], 1e-10f);
    float pt = target_probs[roff];
    float u  = uniform_samples[idx];
    shAcc[tid] = ((u < fminf(1.0f, pt / pd)) && (tid < n_b)) ? 1 : 0;
    shTok[tid] = dtok;
  }
  __syncthreads();

  if (tid == 0) {
    int na = 0;
    while (na < n_b && shAcc[na]) ++na;            // leading accepted run
    const int all = (na == n_b) ? 1 : 0;
    int rp = na; rp = rp > K - 1 ? K - 1 : rp; rp = rp < 0 ? 0 : rp;
    int ri = start + rp; ri = ri < 0 ? 0 : (ri > T - 1 ? T - 1 : ri);
    sh_meta[0] = start; sh_meta[1] = na; sh_meta[2] = all; sh_meta[3] = ri;
  }
  __syncthreads();
  const int num_acc = sh_meta[1];
  const int all_acc = sh_meta[2];

  int recovered = 0;
  if (!all_acc) {
    // Block-uniform row index; readfirstlane so descriptor math is scalar.
    const int rrow = __builtin_amdgcn_readfirstlane(sh_meta[3]);
    const float* tRow = target_probs + (size_t)rrow * (size_t)V;
    const float* dRow = draft_probs  + (size_t)rrow * (size_t)V;
    const int ntiles = (V + TILE - 1) / TILE;

    const unsigned ldsT[2] = { (unsigned)(uintptr_t)(void*)&bufT[0][0],
                               (unsigned)(uintptr_t)(void*)&bufT[1][0] };
    const unsigned ldsD[2] = { (unsigned)(uintptr_t)(void*)&bufD[0][0],
                               (unsigned)(uintptr_t)(void*)&bufD[1][0] };

    // zero per-tile accumulators (visible after first in-loop barrier)
    if (tid < MAXTILES) { tileA[tid] = 0.f; tileT[tid] = 0.f; }

    // ---- Pass A: TDM-staged streaming, double buffered ----
    if (tid < 32) {  // wave 0 only: TDM ignores EXEC, so gate at wave granularity
      tdm_load_tile(tRow, ldsT[0], (unsigned)V, TILE);
      tdm_load_tile(dRow, ldsD[0], (unsigned)V, TILE);
    }
    for (int tile = 0; tile < ntiles; ++tile) {
      const int pb = tile & 1;
      if (tid < 32) {
        const int nt = tile + 1;
        if (nt < ntiles) {
          const unsigned rem = (unsigned)(V - nt * TILE);
          tdm_load_tile(tRow + (size_t)nt * TILE, ldsT[nt & 1], rem, TILE);
          tdm_load_tile(dRow + (size_t)nt * TILE, ldsD[nt & 1], rem, TILE);
          __builtin_amdgcn_s_wait_tensorcnt(2);  // oldest pair (this tile) done
        } else {
          __builtin_amdgcn_s_wait_tensorcnt(0);
        }
      }
      __syncthreads();  // DMA for this tile visible to all waves

      // lane-consecutive LDS reads: conflict-free across 64 banks
      float sA = 0.f, sT = 0.f;
      const int base = tile * TILE;
      #pragma unroll
      for (int j = 0; j < EPT; ++j) {
        const int li = tid + j * NTHREADS;
        float tv = bufT[pb][li];
        float dv = bufD[pb][li];
        if (base + li >= V) { tv = 0.f; dv = 0.f; }
        sA += fmaxf(tv - dv, 0.f);
        sT += tv;
      }
      // wave32 cross-lane reduce, then one LDS float-atomic per wave
      sA = wave_reduce_add(sA);
      sT = wave_reduce_add(sT);
      if (lane == 0) {
        atomicAdd(&tileA[tile], sA);
        atomicAdd(&tileT[tile], sT);
      }
      __syncthreads();  // reads done -> buffer reusable by next TDM issue
    }

    // ---- find crossing tile from tile sums ----
    if (tid == 0) {
      float totalA = 0.f;
      for (int i = 0; i < ntiles; ++i) totalA += tileA[i];
      const int mode = (totalA > 1e-10f) ? 1 : 0;  // 1: adjusted/s, 0: fallback t_row
      const float u = resample_uniforms[b];
      const float thr = mode ? u * totalA : u;      // cdf<u  <=>  cumsum < thr
      float run = 0.f; int cross = -1; float E = 0.f;
      for (int i = 0; i < ntiles; ++i) {
        const float s = mode ? tileA[i] : tileT[i];
        if (run + s >= thr) { cross = i; E = run; break; }
        run += s;
      }
      sh_meta[4] = cross; sh_meta[5] = mode;
      sh_fmeta[0] = thr;  sh_fmeta[1] = E;
    }
    __syncthreads();

    const int cross = sh_meta[4];
    if (cross < 0) {
      recovered = V - 1;  // thr beyond total (u ~ 1 FP edge) -> clamp like reference
    } else {
      const int   mode = sh_meta[5];
      const float thr  = sh_fmeta[0];
      const float E    = sh_fmeta[1];

      // ---- Pass B: re-read only the crossing tile (L2-hot), exact index ----
      const int base = cross * TILE + tid * EPT;   // contiguous 32B per thread
      float vals[EPT];
      float sub = 0.f;
      #pragma unroll
      for (int j = 0; j < EPT; ++j) {
        const int e = base + j;
        float x = 0.f;
        if (e < V) {
          const float tv = tRow[e];
          x = mode ? fmaxf(tv - dRow[e], 0.f) : tv;
        }
        vals[j] = x; sub += x;
      }
      // wave32 inclusive scan of sub-chunk sums
      float isc = sub;
      #pragma unroll
      for (int off = 1; off < 32; off <<= 1) {
        const float y = __shfl_up(isc, off, 32);
        if (lane >= off) isc += y;
      }
      if (lane == 31) sh_wsum[wid] = isc;          // per-wave total
      __syncthreads();
      if (tid == 0) {                               // exclusive scan of 16 wave totals
        float acc = 0.f;
        for (int w = 0; w < NWAVES; ++w) { const float t = sh_wsum[w]; sh_woff[w] = acc; acc += t; }
      }
      __syncthreads();
      const float incl = E + sh_woff[wid] + isc;
      const float excl = incl - sub;
      int cnt = 0;
      if (incl < thr) {
        cnt = EPT;                       // whole sub-chunk below threshold
      } else if (excl < thr) {
        float run = excl;                // crossing lives in this sub-chunk
        #pragma unroll
        for (int j = 0; j < EPT; ++j) {
          run += vals[j];
          if (run < thr) ++cnt;
        }
      }
      cnt = wave_reduce_add_i(cnt);
      if (lane == 0 && cnt != 0) atomicAdd(&sh_count, cnt);
      __syncthreads();
      recovered = cross * TILE + sh_count;
      if (recovered > V - 1) recovered = V - 1;
    }
  }

  // ---- assemble outputs ----
  const int final_tok = all_acc ? bonus_token_ids[b] : recovered;
  if (tid <= K) {
    int val;
    if (tid < num_acc)       val = shTok[tid];
    else if (tid == num_acc) val = final_tok;
    else                     val = -1;
    outbuf[(size_t)b * (K + 1) + tid] = val;
  }
  if (tid == 0) {
    int* num_accepted     = outbuf + (size_t)B * (K + 1);
    int* accepted_counts  = num_accepted + B;
    int* recovered_counts = accepted_counts + B;
    int* bonus_counts     = recovered_counts + B;
    num_accepted[b]     = num_acc + 1;
    accepted_counts[b]  = num_acc;
    recovered_counts[b] = all_acc ? 0 : 1;
    bonus_counts[b]     = all_acc ? 1 : 0;
  }
}

extern "C" void kernel_launch(void* const* d_in, const int* in_sizes, int n_in,
                              void* d_out, int out_size, void* d_ws, size_t ws_size,
                              hipStream_t stream) {
  (void)n_in; (void)out_size; (void)d_ws; (void)ws_size;
  const int*   draft_token_ids   = (const int*)d_in[0];
  const int*   num_draft_tokens  = (const int*)d_in[1];
  const float* draft_probs       = (const float*)d_in[2];
  const float* target_probs      = (const float*)d_in[3];
  const int*   bonus_token_ids   = (const int*)d_in[4];
  const float* uniform_samples   = (const float*)d_in[5];
  const float* resample_uniforms = (const float*)d_in[6];

  const int T = in_sizes[0];
  const int B = in_sizes[1];
  const int V = (int)((long long)in_sizes[2] / (long long)T);
  const int K = T / B;

  rejection_sampler_kernel<<<B, NTHREADS, 0, stream>>>(
      draft_token_ids, num_draft_tokens, draft_probs, target_probs,
      bonus_token_ids, uniform_samples, resample_uniforms,
      (int*)d_out, B, K, V, T);
}